// Sylvester_ortho_VAE_47364899340857
// MI455X (gfx1250) — compile-verified
//
#include <hip/hip_runtime.h>
#include <hip/hip_bf16.h>
#include <cstddef>

typedef float v2f __attribute__((ext_vector_type(2)));
typedef float v8f __attribute__((ext_vector_type(8)));

#define BB 32          // batch
#define ZDIM 256
#define MDIM 64
#define KFLOW 8
#define NEWTON_STEPS 100

// ---------------------------------------------------------------------------
// Activation helper
// ---------------------------------------------------------------------------
__device__ __forceinline__ float act_apply(float v, int act) {
  if (act == 1) return tanhf(v);
  if (act == 2) {
    float sp = (v > 20.f) ? v : log1pf(expf(v));
    return fminf(fmaxf(sp, 1e-4f), 5.f);
  }
  return v;
}

// ---------------------------------------------------------------------------
// Streaming projection: out[b][j] = act( sum_e A[b][e] * W[e][j] + bias[j] )
// A is [32][E], W is [E][N]. Each thread owns one column j with 32
// accumulators; W is read exactly once (bandwidth-bound by design), A is
// staged through LDS in 512-wide chunks.
// ---------------------------------------------------------------------------
__global__ __launch_bounds__(256) void proj_kernel(
    const float* __restrict__ A, const float* __restrict__ W,
    const float* __restrict__ bias, float* __restrict__ out,
    int E, int N, int act) {
  __shared__ float sA[BB][512];
  const int j = blockIdx.x * 256 + threadIdx.x;

  float acc[BB];
#pragma unroll
  for (int b = 0; b < BB; ++b) acc[b] = 0.f;

  for (int e0 = 0; e0 < E; e0 += 512) {
    const int chunk = min(512, E - e0);
    for (int idx = threadIdx.x; idx < BB * chunk; idx += 256) {
      const int b = idx / chunk, e = idx % chunk;
      sA[b][e] = A[b * E + e0 + e];
    }
    __syncthreads();
    if (j < N) {
      for (int e = 0; e < chunk; ++e) {
        const size_t pe = (size_t)(e0 + e + 16);
        if (pe < (size_t)E)  // speculative prefetch ahead (global_prefetch_b8)
          __builtin_prefetch(W + pe * (size_t)N + j, 0, 1);
        const float w = W[(size_t)(e0 + e) * (size_t)N + j];
#pragma unroll
        for (int b = 0; b < BB; ++b) acc[b] = fmaf(sA[b][e], w, acc[b]);
      }
    }
    __syncthreads();
  }

  if (j < N) {
    const float bv = bias ? bias[j] : 0.f;
#pragma unroll
    for (int b = 0; b < BB; ++b)
      out[(size_t)b * (size_t)N + j] = act_apply(acc[b] + bv, act);
  }
}

// ---------------------------------------------------------------------------
// Newton-Schulz orthogonalization, 100 steps, fully LDS-resident.
// One workgroup (256 threads = 8 wave32) per n in [0,256).
// Qc: [Z=256][M=64] fp32.  Per step:
//   phase 1: S = Qc^T Qc via V_WMMA_F32_16X16X4_F32; T = 1.5 I - 0.5 S -> LDS
//   phase 2: Qc[z-rows of this wave] = Qc[rows] @ T  (in place, per-wave rows)
// ---------------------------------------------------------------------------
__global__ __launch_bounds__(256) void ortho_kernel(float* __restrict__ Q) {
  __shared__ float sQ[ZDIM][MDIM + 1];   // +1 pad: stride 65 kills bank conflicts
  __shared__ float sT[MDIM][MDIM + 1];
  __shared__ float red[256];

  const int n = blockIdx.x;
  const int t = threadIdx.x;
  const int wave = t >> 5;           // 8 waves (wave32)
  const int lane = t & 31;
  const int lm   = lane & 15;        // fragment row/col within 16
  const int koff = (lane < 16) ? 0 : 2;
  const int hi8  = (lane < 16) ? 0 : 8;

  // ---- load row t of Qc, accumulate sum of squares ----
  const float* src = Q + (size_t)n * (ZDIM * MDIM) + (size_t)t * MDIM;
  float ss = 0.f;
  for (int c = 0; c < MDIM; c += 4) {
    const float4 v = *(const float4*)(src + c);
    sQ[t][c] = v.x; sQ[t][c + 1] = v.y; sQ[t][c + 2] = v.z; sQ[t][c + 3] = v.w;
    ss += v.x * v.x + v.y * v.y + v.z * v.z + v.w * v.w;
  }
  red[t] = ss;
  __syncthreads();
  for (int off = 128; off > 0; off >>= 1) {
    if (t < off) red[t] += red[t + off];
    __syncthreads();
  }
  const float rinv = 1.f / sqrtf(red[0]);
  for (int c = 0; c < MDIM; ++c) sQ[t][c] *= rinv;
  __syncthreads();

#pragma unroll 1
  for (int step = 0; step < NEWTON_STEPS; ++step) {
    // ---------------- phase 1: S tiles -> T = 1.5I - 0.5S ----------------
#pragma unroll
    for (int tt = 0; tt < 2; ++tt) {
      const int tile = wave * 2 + tt;
      const int ti = tile >> 2, tj = tile & 3;
      v8f acc = {0.f, 0.f, 0.f, 0.f, 0.f, 0.f, 0.f, 0.f};
      for (int kz = 0; kz < ZDIM; kz += 4) {
        v2f a, b;
        a.x = sQ[kz + koff][ti * 16 + lm];
        a.y = sQ[kz + koff + 1][ti * 16 + lm];
        b.x = sQ[kz + koff][tj * 16 + lm];
        b.y = sQ[kz + koff + 1][tj * 16 + lm];
        acc = __builtin_amdgcn_wmma_f32_16x16x4_f32(
            false, a, false, b, (short)0, acc, false, false);
      }
#pragma unroll
      for (int i = 0; i < 8; ++i) {
        const int mrow = ti * 16 + hi8 + i;
        const int ncol = tj * 16 + lm;
        sT[mrow][ncol] = ((mrow == ncol) ? 1.5f : 0.f) - 0.5f * acc[i];
      }
    }
    __syncthreads();

    // ---------------- phase 2: Qc[rows] = Qc[rows] @ T (in place) ----------
#pragma unroll
    for (int q = 0; q < 2; ++q) {
      const int zt = wave * 2 + q;            // this wave's z-tile
      v8f acc[4];
#pragma unroll
      for (int mt = 0; mt < 4; ++mt)
        acc[mt] = (v8f){0.f, 0.f, 0.f, 0.f, 0.f, 0.f, 0.f, 0.f};
      for (int kk = 0; kk < MDIM; kk += 4) {
        v2f a;
        a.x = sQ[zt * 16 + lm][kk + koff];
        a.y = sQ[zt * 16 + lm][kk + koff + 1];
#pragma unroll
        for (int mt = 0; mt < 4; ++mt) {
          v2f b;
          b.x = sT[kk + koff][mt * 16 + lm];
          b.y = sT[kk + koff + 1][mt * 16 + lm];
          acc[mt] = __builtin_amdgcn_wmma_f32_16x16x4_f32(
              false, a, false, b, (short)0, acc[mt], false, false);
        }
      }
#pragma unroll
      for (int mt = 0; mt < 4; ++mt)
#pragma unroll
        for (int i = 0; i < 8; ++i)
          sQ[zt * 16 + hi8 + i][mt * 16 + lm] = acc[mt][i];
    }
    __syncthreads();
  }

  // ---- write back ----
  float* dst = Q + (size_t)n * (ZDIM * MDIM) + (size_t)t * MDIM;
  for (int c = 0; c < MDIM; c += 4) {
    float4 v;
    v.x = sQ[t][c]; v.y = sQ[t][c + 1]; v.z = sQ[t][c + 2]; v.w = sQ[t][c + 3];
    *(float4*)(dst + c) = v;
  }
}

// ---------------------------------------------------------------------------
// Sylvester flow, one block per batch row.  Uses the algebraic collapse:
//   u = q^T z ; pre = Rt~ u + bf ; hk = tanh(pre) ; v = R~ hk ; z += q v
// where R~ / Rt~ are strict-upper + (d1/d2) diagonal.
// ---------------------------------------------------------------------------
__global__ __launch_bounds__(256) void flow_kernel(
    const float* __restrict__ QO,    // [B*K][Z][M]
    const float* __restrict__ Rraw,  // [B][M][M][K]
    const float* __restrict__ Rtraw, // [B][M][M][K]
    const float* __restrict__ d1,    // [B][M][K] (tanh'd)
    const float* __restrict__ d2,    // [B][M][K] (tanh'd)
    const float* __restrict__ bf,    // [B][M][K]
    const float* __restrict__ eps,   // [B][Z]
    const float* __restrict__ mu,    // [B][Z]
    const float* __restrict__ var,   // [B][Z]
    float* __restrict__ out_ldj, float* __restrict__ out_z0,
    float* __restrict__ out_z) {
  __shared__ float sz[ZDIM];
  __shared__ float su[MDIM];
  __shared__ float shk[MDIM];
  __shared__ float sv[MDIM];
  __shared__ float sred[256];

  const int b = blockIdx.x;
  const int t = threadIdx.x;

  const float z0 = eps[b * ZDIM + t] * sqrtf(var[b * ZDIM + t]) + mu[b * ZDIM + t];
  sz[t] = z0;
  out_z0[b * ZDIM + t] = z0;
  float ldj_acc = 0.f;
  __syncthreads();

  for (int k = 0; k < KFLOW; ++k) {
    const float* q = QO + (size_t)(b * KFLOW + k) * (ZDIM * MDIM);

    // u[m] = sum_z q[z][m] * z[z]
    if (t < MDIM) {
      float u = 0.f;
      for (int z = 0; z < ZDIM; ++z) u = fmaf(q[z * MDIM + t], sz[z], u);
      su[t] = u;
    }
    __syncthreads();

    // hk[m^] = tanh( sum_{m>m^} Rt[m^][m] u[m] + d2[m^] u[m^] + bf[m^] )
    if (t < MDIM) {
      const float* rt_row = Rtraw + ((size_t)b * MDIM + t) * (MDIM * KFLOW) + k;
      float pre = 0.f;
      for (int m = t + 1; m < MDIM; ++m) pre = fmaf(rt_row[m * KFLOW], su[m], pre);
      const float d2v = d2[(b * MDIM + t) * KFLOW + k];
      pre += d2v * su[t] + bf[(b * MDIM + t) * KFLOW + k];
      shk[t] = tanhf(pre);
    }
    __syncthreads();

    // v[m] = sum_{n>m} R[m][n] hk[n] + d1[m] hk[m] ; ldj contribution
    if (t < MDIM) {
      const float* r_row = Rraw + ((size_t)b * MDIM + t) * (MDIM * KFLOW) + k;
      float v = 0.f;
      for (int nn = t + 1; nn < MDIM; ++nn) v = fmaf(r_row[nn * KFLOW], shk[nn], v);
      const float d1v = d1[(b * MDIM + t) * KFLOW + k];
      const float d2v = d2[(b * MDIM + t) * KFLOW + k];
      const float hkt = shk[t];
      sv[t] = v + d1v * hkt;
      const float diag = 1.f + (1.f - hkt * hkt) * d1v * d2v;
      sred[t] = logf(fabsf(diag));
    } else {
      sred[t] = 0.f;
    }
    __syncthreads();

    // z[z] += sum_m q[z][m] v[m]
    {
      const float* qrow = q + (size_t)t * MDIM;
      float add = 0.f;
#pragma unroll 8
      for (int m = 0; m < MDIM; ++m) add = fmaf(qrow[m], sv[m], add);
      sz[t] += add;
    }
    __syncthreads();

    for (int off = 128; off > 0; off >>= 1) {
      if (t < off) sred[t] += sred[t + off];
      __syncthreads();
    }
    if (t == 0) ldj_acc += sred[0];
    __syncthreads();
  }

  out_z[b * ZDIM + t] = sz[t];
  if (t == 0) out_ldj[b] = ldj_acc;
}

// ---------------------------------------------------------------------------
extern "C" void kernel_launch(void* const* d_in, const int* in_sizes, int n_in,
                              void* d_out, int out_size, void* d_ws, size_t ws_size,
                              hipStream_t stream) {
  (void)in_sizes; (void)n_in; (void)out_size; (void)ws_size;
  const float* x     = (const float*)d_in[0];
  const float* eps   = (const float*)d_in[1];
  const float* enc_W = (const float*)d_in[2];
  const float* enc_b = (const float*)d_in[3];
  const float* mu_W  = (const float*)d_in[4];
  const float* mu_b  = (const float*)d_in[5];
  const float* var_W = (const float*)d_in[6];
  const float* var_b = (const float*)d_in[7];
  const float* q_W   = (const float*)d_in[8];
  const float* q_b   = (const float*)d_in[9];
  const float* bf_W  = (const float*)d_in[10];
  const float* bf_b  = (const float*)d_in[11];
  const float* r_W   = (const float*)d_in[12];
  const float* r_b   = (const float*)d_in[13];
  const float* rt_W  = (const float*)d_in[14];
  const float* rt_b  = (const float*)d_in[15];
  const float* d1_W  = (const float*)d_in[16];
  const float* d1_b  = (const float*)d_in[17];
  const float* d2_W  = (const float*)d_in[18];
  const float* d2_b  = (const float*)d_in[19];
  const float* dec_W = (const float*)d_in[20];
  const float* dec_b = (const float*)d_in[21];
  const float* pmu_W = (const float*)d_in[22];
  const float* pmu_b = (const float*)d_in[23];

  const int DIN = 3072, ENC = 512, DEC = 1024;
  const int NQ  = KFLOW * ZDIM * MDIM;   // 131072
  const int NR  = KFLOW * MDIM * MDIM;   // 32768
  const int NKM = KFLOW * MDIM;          // 512

  // output layout: x_mean | mu | var | ldj | z0 | z
  float* out      = (float*)d_out;
  float* o_xmean  = out;
  float* o_mu     = o_xmean + BB * DIN;      // 98304
  float* o_var    = o_mu + BB * ZDIM;
  float* o_ldj    = o_var + BB * ZDIM;
  float* o_z0     = o_ldj + BB;
  float* o_z      = o_z0 + BB * ZDIM;

  // workspace layout
  float* ws   = (float*)d_ws;
  float* w_h  = ws;                          // 32*512
  float* w_Q  = w_h + BB * ENC;              // 32*131072
  float* w_R  = w_Q + (size_t)BB * NQ;       // 32*32768
  float* w_Rt = w_R + (size_t)BB * NR;
  float* w_bf = w_Rt + (size_t)BB * NR;
  float* w_d1 = w_bf + BB * NKM;
  float* w_d2 = w_d1 + BB * NKM;
  float* w_hd = w_d2 + BB * NKM;             // 32*1024

  auto blocks = [](int N) { return dim3((unsigned)((N + 255) / 256)); };

  // encoder + all h-projections (weight streaming, read each W once)
  proj_kernel<<<blocks(ENC), 256, 0, stream>>>(x,   enc_W, enc_b, w_h,  DIN, ENC, 1);
  proj_kernel<<<blocks(ZDIM), 256, 0, stream>>>(w_h, mu_W,  mu_b,  o_mu, ENC, ZDIM, 0);
  proj_kernel<<<blocks(ZDIM), 256, 0, stream>>>(w_h, var_W, var_b, o_var, ENC, ZDIM, 2);
  proj_kernel<<<blocks(NQ),  256, 0, stream>>>(w_h, q_W,   q_b,   w_Q,  ENC, NQ, 0);
  proj_kernel<<<blocks(NKM), 256, 0, stream>>>(w_h, bf_W,  bf_b,  w_bf, ENC, NKM, 0);
  proj_kernel<<<blocks(NR),  256, 0, stream>>>(w_h, r_W,   r_b,   w_R,  ENC, NR, 0);
  proj_kernel<<<blocks(NR),  256, 0, stream>>>(w_h, rt_W,  rt_b,  w_Rt, ENC, NR, 0);
  proj_kernel<<<blocks(NKM), 256, 0, stream>>>(w_h, d1_W,  d1_b,  w_d1, ENC, NKM, 1);
  proj_kernel<<<blocks(NKM), 256, 0, stream>>>(w_h, d2_W,  d2_b,  w_d2, ENC, NKM, 1);

  // 100-step Newton orthogonalization: 256 groups, WMMA f32, LDS resident
  ortho_kernel<<<dim3(256), 256, 0, stream>>>(w_Q);

  // Sylvester flow (writes ldj, z0, z)
  flow_kernel<<<dim3(BB), 256, 0, stream>>>(w_Q, w_R, w_Rt, w_d1, w_d2, w_bf,
                                            eps, o_mu, o_var, o_ldj, o_z0, o_z);

  // decoder
  proj_kernel<<<blocks(DEC), 256, 0, stream>>>(o_z,  dec_W, dec_b, w_hd,    ZDIM, DEC, 1);
  proj_kernel<<<blocks(DIN), 256, 0, stream>>>(w_hd, pmu_W, pmu_b, o_xmean, DEC,  DIN, 0);
}